// EMAVQ_14740327760385
// MI455X (gfx1250) — compile-verified
//
#include <hip/hip_runtime.h>

typedef __attribute__((ext_vector_type(2))) float v2f;
typedef __attribute__((ext_vector_type(8))) float v8f;

#define VOCAB 4096
#define DECAY 0.99f
#define OMD   0.01f
#define EPSV  1e-5f

// ---------------------------------------------------------------------------
// ws layout (floats): [0, V) counts | [V, 4V) sums(V,3) | [4V] sse | [4V+1] nsum
// ---------------------------------------------------------------------------

__global__ void vq_init(float* __restrict__ ws, int n) {
    int i = blockIdx.x * blockDim.x + threadIdx.x;
    if (i < n) ws[i] = 0.0f;
}

// One wave32 handles one tile of 16 points; loops over 256 tiles of 16 codes.
// WMMA computes D[code][point] = -2*f.e + |e|^2 for a 16x16 tile.
__global__ __launch_bounds__(256)
void vq_assign(const float* __restrict__ feats,
               const float* __restrict__ embed,
               float* __restrict__ quant_out,   // (N,3)
               float* __restrict__ idx_out,     // (N) as float
               float* __restrict__ counts,      // (V)
               float* __restrict__ sums,        // (V,3)
               float* __restrict__ sse)         // scalar accumulator
{
    __shared__ float ldsA[VOCAB * 4];   // 64 KB: per code (-2ex,-2ey,-2ez,|e|^2)

    for (int c = threadIdx.x; c < VOCAB; c += blockDim.x) {
        float ex = embed[c * 3 + 0];
        float ey = embed[c * 3 + 1];
        float ez = embed[c * 3 + 2];
        ldsA[c * 4 + 0] = -2.0f * ex;
        ldsA[c * 4 + 1] = -2.0f * ey;
        ldsA[c * 4 + 2] = -2.0f * ez;
        ldsA[c * 4 + 3] = ex * ex + ey * ey + ez * ez;
    }
    __syncthreads();

    const int lane = threadIdx.x & 31;
    const int wave = threadIdx.x >> 5;
    const int tile = blockIdx.x * 8 + wave;      // 16 points per tile
    const int p    = tile * 16 + (lane & 15);    // this lane's point
    const bool hi  = lane >= 16;

    // B operand (points), loop invariant.
    // lanes 0-15: (K0,K1) = (fx, fy); lanes 16-31: (K2,K3) = (fz, 1)
    v2f b;
    if (!hi) { b.x = feats[p * 3 + 0]; b.y = feats[p * 3 + 1]; }
    else     { b.x = feats[p * 3 + 2]; b.y = 1.0f; }

    float mind = 3.4e38f;
    int   mini = 0;

    for (int ct = 0; ct < VOCAB / 16; ++ct) {
        // A operand (codes): lane L holds code row ct*16 + (L&15);
        // lanes 0-15 carry K=0,1; lanes 16-31 carry K=2,3.
        const int c = ct * 16 + (lane & 15);
        const float* ap = &ldsA[c * 4 + (hi ? 2 : 0)];
        v2f a;
        a.x = ap[0];
        a.y = ap[1];

        v8f d = {};
        d = __builtin_amdgcn_wmma_f32_16x16x4_f32(
                /*neg_a=*/false, a, /*neg_b=*/false, b,
                /*c_mod=*/(short)0, d, /*reuse_a=*/false, /*reuse_b=*/false);

        // D layout: lane(N=point), VGPR r = code row (hi half: rows 8-15)
        const int cbase = ct * 16 + (hi ? 8 : 0);
#pragma unroll
        for (int r = 0; r < 8; ++r) {
            float dv = d[r];
            if (dv < mind) { mind = dv; mini = cbase + r; }
        }
    }

    // Merge the two half-wave code ranges for each point.
    float od = __shfl_xor(mind, 16);
    int   oi = __shfl_xor(mini, 16);
    if (od < mind || (od == mind && oi < mini)) { mind = od; mini = oi; }

    float e2 = 0.0f;
    if (!hi) {
        idx_out[p] = (float)mini;
        float qx = embed[mini * 3 + 0];
        float qy = embed[mini * 3 + 1];
        float qz = embed[mini * 3 + 2];
        quant_out[p * 3 + 0] = qx;
        quant_out[p * 3 + 1] = qy;
        quant_out[p * 3 + 2] = qz;
        float fx = feats[p * 3 + 0];
        float fy = feats[p * 3 + 1];
        float fz = feats[p * 3 + 2];
        float dx = qx - fx, dy = qy - fy, dz = qz - fz;
        e2 = dx * dx + dy * dy + dz * dz;
        atomicAdd(&counts[mini], 1.0f);
        atomicAdd(&sums[mini * 3 + 0], fx);
        atomicAdd(&sums[mini * 3 + 1], fy);
        atomicAdd(&sums[mini * 3 + 2], fz);
    }
    // reduce e2 over the low 16 lanes (hi lanes carry 0)
    e2 += __shfl_xor(e2, 8);
    e2 += __shfl_xor(e2, 4);
    e2 += __shfl_xor(e2, 2);
    e2 += __shfl_xor(e2, 1);
    if (lane == 0) atomicAdd(sse, e2);
}

__global__ void vq_update(const float* __restrict__ ema_cs,
                          const float* __restrict__ ema_w,
                          const float* __restrict__ counts,
                          const float* __restrict__ sums,
                          float* __restrict__ out_ncs,
                          float* __restrict__ out_new_w,
                          float* __restrict__ nsum)
{
    int v = blockIdx.x * blockDim.x + threadIdx.x;
    if (v >= VOCAB) return;
    float ncs = DECAY * ema_cs[v] + OMD * counts[v];
    out_ncs[v] = ncs;
#pragma unroll
    for (int d = 0; d < 3; ++d)
        out_new_w[v * 3 + d] = DECAY * ema_w[v * 3 + d] + OMD * sums[v * 3 + d];
    atomicAdd(nsum, ncs);
}

__global__ void vq_final(const float* __restrict__ out_ncs,
                         const float* __restrict__ out_new_w,
                         const float* __restrict__ nsum,
                         const float* __restrict__ sse,
                         float* __restrict__ out_embed,
                         float* __restrict__ out_loss,
                         float inv_count)
{
    int v = blockIdx.x * blockDim.x + threadIdx.x;
    if (v == 0) *out_loss = 1.25f * (*sse) * inv_count;  // both MSE terms equal fwd
    if (v >= VOCAB) return;
    float n  = *nsum;
    float cs = (out_ncs[v] + EPSV) / (n + (float)VOCAB * EPSV) * n;
    float inv = 1.0f / cs;
#pragma unroll
    for (int d = 0; d < 3; ++d)
        out_embed[v * 3 + d] = out_new_w[v * 3 + d] * inv;
}

extern "C" void kernel_launch(void* const* d_in, const int* in_sizes, int n_in,
                              void* d_out, int out_size, void* d_ws, size_t ws_size,
                              hipStream_t stream) {
    const float* feats  = (const float*)d_in[0];   // (4,8192,3)
    const float* embed  = (const float*)d_in[1];   // (4096,3)
    const float* ema_cs = (const float*)d_in[2];   // (4096,)
    const float* ema_w  = (const float*)d_in[3];   // (4096,3)

    const int N = in_sizes[0] / 3;                 // 32768 points

    float* out        = (float*)d_out;
    float* quant_out  = out;                       // N*3
    float* idx_out    = quant_out + (size_t)N * 3; // N
    float* loss_out   = idx_out + N;               // 1
    float* ncs_out    = loss_out + 1;              // V
    float* neww_out   = ncs_out + VOCAB;           // V*3
    float* nembed_out = neww_out + VOCAB * 3;      // V*3

    float* ws     = (float*)d_ws;
    float* counts = ws;                // V
    float* sums   = ws + VOCAB;        // V*3
    float* sseacc = ws + VOCAB * 4;    // 1
    float* nsum   = sseacc + 1;        // 1

    const int ws_floats = VOCAB * 4 + 2;
    vq_init<<<(ws_floats + 255) / 256, 256, 0, stream>>>(ws, ws_floats);

    const int tiles  = N / 16;         // 2048 point tiles
    const int blocks = tiles / 8;      // 8 wave32 per 256-thread block
    vq_assign<<<blocks, 256, 0, stream>>>(feats, embed, quant_out, idx_out,
                                          counts, sums, sseacc);

    vq_update<<<VOCAB / 256, 256, 0, stream>>>(ema_cs, ema_w, counts, sums,
                                               ncs_out, neww_out, nsum);

    vq_final<<<VOCAB / 256, 256, 0, stream>>>(ncs_out, neww_out, nsum, sseacc,
                                              nembed_out, loss_out,
                                              1.0f / (float)(N * 3));
}